// Morphology_7593502179355
// MI455X (gfx1250) — compile-verified
//
#include <hip/hip_runtime.h>
#include <hip/hip_bf16.h>
#include <stdint.h>

// Morphological dilation2d 7x7 (max-plus) + ReLU, fp32.
// x: (16,96,128,128), weight: (96,7,7), out: (16,96,128,128).
//
// CDNA5 data path: halo tile staged into LDS with
// GLOBAL_LOAD_ASYNC_TO_LDS_B128 (ASYNCcnt-tracked async copy,
// CDNA5 ISA). Async loads are issued FIRST and their
// latency overlapped with zeroing of the (disjoint) LDS halo; one
// s_wait_asynccnt + barrier precedes compute. Compute uses 4x4 register
// blocking: 10x10 LDS footprint per thread read as 10 rows x 3 ds_load_b128,
// results written as 4 global_store_b128.

#define B_   16
#define C_   96
#define H_   128
#define W_   128
#define KS   7
#define PADS 3
#define TH   64                 // output rows per block
#define LROW (TH + KS - 1)      // 70 LDS rows (with halo)
#define LSTR 136                // LDS row stride (floats); interior at col 4 -> 16B aligned
#define NTHREADS 512            // 16 waves

__global__ __launch_bounds__(NTHREADS)
void morph7x7_relu_kernel(const float* __restrict__ x,
                          const float* __restrict__ weight,
                          float* __restrict__ out) {
    __shared__ __align__(16) float tile[LROW * LSTR];   // 38080 B

    const int tid  = threadIdx.x;
    const int c    = blockIdx.y;
    const int b    = blockIdx.z;
    const int h0   = blockIdx.x * TH;

    const float* __restrict__ xplane =
        x + ((size_t)(b * C_ + c)) * (size_t)(H_ * W_);

    // ---- 1) issue async copies of all in-range rows into the LDS interior ----
    // One B128 per lane, one full 512B row per wave issue; rows are
    // wave-uniform so the guard does not diverge.
    const int wave = tid >> 5;
    const int lane = tid & 31;
    for (int lr = wave; lr < LROW; lr += NTHREADS / 32) {
        const int gr = h0 - PADS + lr;
        if (gr >= 0 && gr < H_) {
            // AS3 byte offset == low 32 bits of the generic pointer
            uint32_t lds_byte =
                (uint32_t)(size_t)(&tile[lr * LSTR + 4]) + (uint32_t)(lane * 16);
            uint64_t gaddr =
                (uint64_t)(xplane + (size_t)gr * W_) + (uint64_t)(lane * 16);
            asm volatile("global_load_async_to_lds_b128 %0, %1, off"
                         :: "v"(lds_byte), "v"(gaddr) : "memory");
        }
    }

    // ---- 2) zero the halo (disjoint from async destinations -> overlaps) ----
    const float4 z4 = make_float4(0.f, 0.f, 0.f, 0.f);
    // 2a: left cols 0..3 and right cols 132..135 of every LDS row
    for (int t = tid; t < LROW * 2; t += NTHREADS) {
        const int r    = t >> 1;
        const int coff = (t & 1) ? (LSTR - 4) : 0;
        *(float4*)&tile[r * LSTR + coff] = z4;
    }
    // 2b: full interior of out-of-range rows (top/bottom image borders)
    for (int t = tid; t < LROW * 32; t += NTHREADS) {
        const int lr  = t >> 5;
        const int seg = t & 31;
        const int gr  = h0 - PADS + lr;
        if (gr < 0 || gr >= H_)
            *(float4*)&tile[lr * LSTR + 4 + seg * 4] = z4;
    }

    // ---- per-channel weights: uniform address -> scalar loads ----
    const float* __restrict__ wc = weight + c * (KS * KS);
    float wr[KS * KS];
    #pragma unroll
    for (int i = 0; i < KS * KS; ++i) wr[i] = wc[i];

    asm volatile("s_wait_asynccnt 0" ::: "memory");
    __syncthreads();

    // ---- 3) compute: 4x4 outputs per thread, 10x10 LDS footprint ----
    const int c0 = (lane) * 4;          // output cols c0..c0+3 (0..127)
    const int r0 = wave * 4;            // output rows (tile-rel) r0..r0+3 (0..63)

    float acc[4][4];
    #pragma unroll
    for (int r = 0; r < 4; ++r)
        #pragma unroll
        for (int s = 0; s < 4; ++s) acc[r][s] = 0.0f;   // folds the final ReLU

    // Input (LDS) rows r0..r0+9; per row, aligned cols c0..c0+11 cover all
    // taps (needed cols are c0+1..c0+10 since ldscol = col + j + 1).
    #pragma unroll
    for (int k = 0; k < 4 + KS - 1 + 3; ++k) {   // k = 0..12 (k >= 10 hits no tap)
        float4 a0 = *(const float4*)&tile[(r0 + k) * LSTR + c0 + 0];
        float4 a1 = *(const float4*)&tile[(r0 + k) * LSTR + c0 + 4];
        float4 a2 = *(const float4*)&tile[(r0 + k) * LSTR + c0 + 8];
        float rowreg[12] = { a0.x, a0.y, a0.z, a0.w,
                             a1.x, a1.y, a1.z, a1.w,
                             a2.x, a2.y, a2.z, a2.w };
        #pragma unroll
        for (int r = 0; r < 4; ++r) {
            const int i = k - r;                 // vertical tap index
            if (i < 0 || i > KS - 1) continue;   // folds at compile time
            #pragma unroll
            for (int j = 0; j < KS; ++j) {
                const float wij = wr[i * KS + j];
                #pragma unroll
                for (int s = 0; s < 4; ++s)
                    acc[r][s] = fmaxf(acc[r][s], rowreg[s + j + 1] + wij);
            }
        }
    }

    float* __restrict__ op =
        out + ((size_t)(b * C_ + c)) * (size_t)(H_ * W_)
            + (size_t)(h0 + r0) * W_ + c0;
    #pragma unroll
    for (int r = 0; r < 4; ++r) {
        float4 v = make_float4(acc[r][0], acc[r][1], acc[r][2], acc[r][3]);
        *(float4*)(op + (size_t)r * W_) = v;
    }
}

extern "C" void kernel_launch(void* const* d_in, const int* in_sizes, int n_in,
                              void* d_out, int out_size, void* d_ws, size_t ws_size,
                              hipStream_t stream) {
    (void)in_sizes; (void)n_in; (void)out_size; (void)d_ws; (void)ws_size;
    const float* x = (const float*)d_in[0];
    const float* w = (const float*)d_in[1];
    float* out     = (float*)d_out;

    dim3 grid(H_ / TH, C_, B_);   // (2, 96, 16)
    morph7x7_relu_kernel<<<grid, NTHREADS, 0, stream>>>(x, w, out);
}